// VectorQuantizer_21638045237923
// MI455X (gfx1250) — compile-verified
//
#include <hip/hip_runtime.h>
#include <stdint.h>

typedef __attribute__((ext_vector_type(2))) float v2f;
typedef __attribute__((ext_vector_type(4))) float v4f;
typedef __attribute__((ext_vector_type(8))) float v8f;

#define EMB_DIM 256          // D
#define KSTEPS  (EMB_DIM/4)  // 64 WMMA K-steps of 4
#define PAD     260          // LDS row stride (dwords): 260 % 64 == 4 -> conflict-free b64 reads
#define NBUF    3            // async pipeline depth (prefetch 2 tiles ahead)

template <int N>
__device__ __forceinline__ void async_wait() {
#if defined(__has_builtin) && __has_builtin(__builtin_amdgcn_s_wait_asynccnt)
    __builtin_amdgcn_s_wait_asynccnt(N);
#else
    asm volatile("s_wait_asynccnt %0" :: "i"(N) : "memory");
#endif
}

// ---------------------------------------------------------------------------
// Kernel 1: per-code squared norms  ||w_k||^2  -> d_ws
// ---------------------------------------------------------------------------
__global__ __launch_bounds__(256) void vq_wnorm_kernel(const float* __restrict__ w,
                                                       float* __restrict__ wnorm) {
    const int r = blockIdx.x * blockDim.x + threadIdx.x;
    const v4f* row = (const v4f*)(w + (size_t)r * EMB_DIM);
    float s = 0.0f;
#pragma unroll 8
    for (int i = 0; i < EMB_DIM / 4; ++i) {
        v4f v = row[i];
        s = fmaf(v.x, v.x, s);
        s = fmaf(v.y, v.y, s);
        s = fmaf(v.z, v.z, s);
        s = fmaf(v.w, v.w, s);
    }
    wnorm[r] = s;
}

// ---------------------------------------------------------------------------
// Kernel 2: fused  argmax_k(||w_k||^2 - 2 x.w_k)  +  gather(weight[idx])
//
// Block = 8 waves; each wave owns a 16-row x tile (A operand pinned in VGPRs).
// One shared 16-code B tile per iteration, staged by async DMA
// (global_load_async_to_lds_b128, ASYNCcnt), triple-buffered so the DMA
// engine never drains; DS reads software-pipelined 4 K-steps ahead.
// ---------------------------------------------------------------------------
__global__ __launch_bounds__(256) void vq_argmax_gather_kernel(
    const float* __restrict__ x, const float* __restrict__ w,
    const float* __restrict__ wnorm, float* __restrict__ out, int ncodes) {

    __shared__ float bt[NBUF][16][PAD];  // 3 x 16x256 weight tiles (+pad) ~= 50 KB
    __shared__ int   s_idx[8][16];       // per-wave winning code index per row

    const int tid  = threadIdx.x;
    const int lane = tid & 31;
    const int wave = tid >> 5;
    const int lo   = lane & 15;          // N col (compute) / M row (A load) position
    const int hi   = lane >> 4;          // K sub-pair selector in A/B operands
    const int rowBase = (blockIdx.x * 8 + wave) * 16;

    // ---- Async stage of one 16x256 B tile into LDS buffer `buf`.
    // 1024 16-byte chunks; 256 threads x 4 chunks. Wave-relative LDS byte
    // offset = low 32 bits of the generic __shared__ pointer (ISA 10.2).
    auto stage_tile = [&](int t, int buf) {
#pragma unroll
        for (int i = 0; i < 4; ++i) {
            const int c = tid + 256 * i;        // chunk id 0..1023
            const int r = c >> 6;               // code row within tile
            const int j = c & 63;               // 16B chunk within row
            uint32_t loff = (uint32_t)(uintptr_t)&bt[buf][r][4 * j];
            const float* gsrc = w + (size_t)(t * 16 + r) * EMB_DIM + 4 * j;
            asm volatile("global_load_async_to_lds_b128 %0, %1, off"
                         :: "v"(loff), "v"(gsrc) : "memory");
        }
    };

    // ---- Preload A operand tile (16x256 fp32) into registers in WMMA layout.
    // A 16x4 f32: lane l holds row l%16; VGPR0/1 = K = 2*(l/16)+{0,1}.
    v2f a[KSTEPS];
    {
        const float* xrow = x + (size_t)(rowBase + lo) * EMB_DIM + 2 * hi;
#pragma unroll
        for (int k = 0; k < KSTEPS; ++k)
            a[k] = *(const v2f*)(xrow + 4 * k);
    }

    float bscore[8];
    int   bidx[8];
#pragma unroll
    for (int r = 0; r < 8; ++r) { bscore[r] = -__builtin_inff(); bidx[r] = 0; }

    const int ntiles = ncodes / 16;

    // Prologue: fill pipeline with tiles 0 and 1.
    stage_tile(0, 0);
    stage_tile(1, 1);
    async_wait<4>();       // tile 0 complete (in-order); tile 1 may be in flight
    __syncthreads();

    int bufc = 0;          // buffer being computed
    int bufp = 2;          // buffer being prefetched (tile t+2)

    for (int t = 0; t < ntiles; ++t) {
        const bool pf = (t + 2 < ntiles);
        if (pf) stage_tile(t + 2, bufp);

        const int col = t * 16 + lo;
        const float wn = wnorm[col];               // issued early, L0-hit

        // ---- B from LDS, software-pipelined 4 K-steps (2 DS loads) ahead.
        const float* bbase = &bt[bufc][lo][2 * hi];
        v2f b0 = *(const v2f*)(bbase + 0);
        v2f b1 = *(const v2f*)(bbase + 4);
        v2f b2 = *(const v2f*)(bbase + 8);
        v2f b3 = *(const v2f*)(bbase + 12);

        v8f acc = {0.f, 0.f, 0.f, 0.f, 0.f, 0.f, 0.f, 0.f};
#pragma unroll
        for (int k = 0; k < KSTEPS; k += 4) {
            v2f n0, n1, n2, n3;
            if (k + 4 < KSTEPS) {                  // compile-time after unroll
                n0 = *(const v2f*)(bbase + 4 * (k + 4));
                n1 = *(const v2f*)(bbase + 4 * (k + 5));
                n2 = *(const v2f*)(bbase + 4 * (k + 6));
                n3 = *(const v2f*)(bbase + 4 * (k + 7));
            }
            acc = __builtin_amdgcn_wmma_f32_16x16x4_f32(false, a[k],     false, b0, (short)0, acc, false, false);
            acc = __builtin_amdgcn_wmma_f32_16x16x4_f32(false, a[k + 1], false, b1, (short)0, acc, false, false);
            acc = __builtin_amdgcn_wmma_f32_16x16x4_f32(false, a[k + 2], false, b2, (short)0, acc, false, false);
            acc = __builtin_amdgcn_wmma_f32_16x16x4_f32(false, a[k + 3], false, b3, (short)0, acc, false, false);
            b0 = n0; b1 = n1; b2 = n2; b3 = n3;
        }

#pragma unroll
        for (int r = 0; r < 8; ++r) {
            float s = fmaf(-2.0f, acc[r], wn);     // ||w||^2 - 2 x.w
            if (s > bscore[r]) { bscore[r] = s; bidx[r] = col; }
        }

        // Allow this iteration's 4 prefetch loads to stay in flight; older
        // (next tile's) loads are complete by in-order ASYNC completion.
        if (pf) async_wait<4>(); else async_wait<0>();
        __syncthreads();

        bufc = (bufc == NBUF - 1) ? 0 : bufc + 1;
        bufp = (bufp == NBUF - 1) ? 0 : bufp + 1;
    }

    // ---- Cross-lane argmax over the 16 column positions (xor masks stay
    // within each 16-lane half). Tie-break to the lower index (jnp.argmax).
#pragma unroll
    for (int m = 1; m <= 8; m <<= 1) {
#pragma unroll
        for (int r = 0; r < 8; ++r) {
            float os = __shfl_xor(bscore[r], m, 32);
            int   oi = __shfl_xor(bidx[r],   m, 32);
            if (os > bscore[r] || (os == bscore[r] && oi < bidx[r])) {
                bscore[r] = os; bidx[r] = oi;
            }
        }
    }

    // Lane 0 holds winners for rows 0..7 ; lane 16 for rows 8..15.
    if (lo == 0) {
#pragma unroll
        for (int r = 0; r < 8; ++r) s_idx[wave][r + 8 * hi] = bidx[r];
    }
    __syncthreads();   // uniform: every wave runs identical trip counts

    // ---- Gather: out[row] = weight[best_idx[row]]  (coalesced float4 copies)
#pragma unroll 4
    for (int m = 0; m < 16; ++m) {
        const int idx = s_idx[wave][m];
        const v4f* src = (const v4f*)(w + (size_t)idx * EMB_DIM);
        v4f* dst = (v4f*)(out + (size_t)(rowBase + m) * EMB_DIM);
#pragma unroll
        for (int it = 0; it < 2; ++it)
            dst[lane + 32 * it] = src[lane + 32 * it];
    }
}

// ---------------------------------------------------------------------------
extern "C" void kernel_launch(void* const* d_in, const int* in_sizes, int n_in,
                              void* d_out, int out_size, void* d_ws, size_t ws_size,
                              hipStream_t stream) {
    const float* x = (const float*)d_in[0];   // (16,1024,256) fp32
    const float* w = (const float*)d_in[1];   // (8192,256) fp32
    float* out     = (float*)d_out;
    float* wnorm   = (float*)d_ws;            // 8192 floats = 32 KB scratch

    const int ntokens = in_sizes[0] / EMB_DIM;   // 16384
    const int ncodes  = in_sizes[1] / EMB_DIM;   // 8192

    vq_wnorm_kernel<<<ncodes / 256, 256, 0, stream>>>(w, wnorm);

    const int blocks = ntokens / 128;            // 8 waves x 16 rows per block
    vq_argmax_gather_kernel<<<blocks, 256, 0, stream>>>(x, w, wnorm, out, ncodes);
}